// KMEANS_67980742361656
// MI455X (gfx1250) — compile-verified
//
#include <hip/hip_runtime.h>

typedef __attribute__((ext_vector_type(2))) float v2f;
typedef __attribute__((ext_vector_type(8))) float v8f;

#define NPTS 65536
#define KCL  1024
#define DDIM 512

// LDS slab: 64 center rows x 64 d-values, row stride 68 floats
// (272B row pitch: keeps 16B stores/loads aligned, spreads the 16 B-fragment
//  rows across banks 4c -> conflict-free ds_load_b64 across all 32 lanes).
#define LDS_STRIDE 68
#define SLAB_FLOATS (64 * LDS_STRIDE)

// ---------------------------------------------------------------------------
// Kernel 1: per-center squared norms c2[k] = sum_d centers[k,d]^2 (+ zero loss)
// ---------------------------------------------------------------------------
__global__ __launch_bounds__(256) void KM_center_norms(
    const float* __restrict__ centers, float* __restrict__ c2,
    float* __restrict__ out) {
  if (blockIdx.x == 0 && threadIdx.x == 0) out[0] = 0.0f;  // loss accumulator
  const int lane = threadIdx.x & 31;
  const int wave = threadIdx.x >> 5;
  const int k = blockIdx.x * 8 + wave;
  const float* c = centers + (size_t)k * DDIM;
  float s = 0.0f;
  #pragma unroll 4
  for (int d = lane; d < DDIM; d += 32) {
    float v = c[d];
    s = fmaf(v, v, s);
  }
  #pragma unroll
  for (int off = 16; off >= 1; off >>= 1) s += __shfl_xor(s, off, 32);
  if (lane == 0) c2[k] = s;
}

// ---------------------------------------------------------------------------
// Kernel 2: argmin over centers via fp32 WMMA with async double-buffered
// centers staging (GLOBAL_LOAD_ASYNC_TO_LDS_B128, ASYNCcnt-tracked).
//
// Block = 256 threads (8 waves); wave w owns rows [row0, row0+16) for ALL
// 1024 centers, so argmin never leaves the wave. 128 chunks linearize
// (16 ktiles x 8 d-chunks); chunk q+1 DMAs into lds[buf^1] while chunk q
// feeds 64 v_wmma_f32_16x16x4_f32 per wave out of lds[buf].
// ---------------------------------------------------------------------------
__global__ __launch_bounds__(256) void KM_argmin_wmma(
    const float* __restrict__ x, const float* __restrict__ centers,
    const float* __restrict__ c2, float* __restrict__ out) {
  __shared__ float ldsB[2][SLAB_FLOATS];

  const int tid  = threadIdx.x;
  const int lane = tid & 31;
  const int wave = tid >> 5;
  const int half = lane >> 4;   // 0: lanes 0-15, 1: lanes 16-31
  const int l15  = lane & 15;
  const int dA   = half * 2;    // fp32 A/B fragment: halves hold d+0..1 / d+2..3

  const int row0 = blockIdx.x * 128 + wave * 16;
  const float* xbase = x + (size_t)(row0 + l15) * DDIM + dA;

  // Async-stage chunk q (ktile = (q>>3)*64, dch = (q&7)*64) into buffer b:
  // 1024 x 16B transfers, 4 per thread, straight to LDS (no VGPR staging).
  auto stage = [&](int q, int b) {
    const int kt  = (q >> 3) << 6;
    const int dch = (q & 7) << 6;
    #pragma unroll
    for (int it = 0; it < 4; ++it) {
      const int idx = it * 256 + tid;
      const int c   = idx >> 4;          // center row within tile
      const int q4  = (idx & 15) * 4;    // float4 column
      const unsigned long long gaddr =
          (unsigned long long)(size_t)(centers + (size_t)(kt + c) * DDIM + dch + q4);
      // Generic shared pointers carry the LDS byte offset in their low 32 bits.
      const unsigned laddr = (unsigned)(size_t)&ldsB[b][c * LDS_STRIDE + q4];
      asm volatile("global_load_async_to_lds_b128 %0, %1, off"
                   :: "v"(laddr), "v"(gaddr) : "memory");
    }
  };

  float best[8];
  int   bidx[8];
  #pragma unroll
  for (int r = 0; r < 8; ++r) { best[r] = 3.4e38f; bidx[r] = 0; }

  const v8f vzero = {0.f, 0.f, 0.f, 0.f, 0.f, 0.f, 0.f, 0.f};
  v8f acc0 = vzero, acc1 = vzero, acc2 = vzero, acc3 = vzero;

  // Prologue: fill buffer 0 with chunk 0.
  stage(0, 0);
  asm volatile("s_wait_asynccnt 0x0" ::: "memory");
  __syncthreads();

  int buf = 0;
  for (int q = 0; q < 128; ++q) {
    const int kt   = (q >> 3) << 6;
    const int dch8 = q & 7;

    if (dch8 == 0) { acc0 = vzero; acc1 = vzero; acc2 = vzero; acc3 = vzero; }

    if (q + 1 < 128) stage(q + 1, buf ^ 1);   // overlap DMA with WMMA

    const float* xb = xbase + (dch8 << 6);
    const float* lb = &ldsB[buf][0];
    #pragma unroll
    for (int dl = 0; dl < 64; dl += 4) {
      v2f a  = *(const v2f*)(xb + dl);
      v2f b0 = *(const v2f*)(lb + (0 * 16 + l15) * LDS_STRIDE + dl + dA);
      v2f b1 = *(const v2f*)(lb + (1 * 16 + l15) * LDS_STRIDE + dl + dA);
      v2f b2 = *(const v2f*)(lb + (2 * 16 + l15) * LDS_STRIDE + dl + dA);
      v2f b3 = *(const v2f*)(lb + (3 * 16 + l15) * LDS_STRIDE + dl + dA);
      acc0 = __builtin_amdgcn_wmma_f32_16x16x4_f32(false, a, false, b0,
                                                   (short)0, acc0, false, false);
      acc1 = __builtin_amdgcn_wmma_f32_16x16x4_f32(false, a, false, b1,
                                                   (short)0, acc1, false, false);
      acc2 = __builtin_amdgcn_wmma_f32_16x16x4_f32(false, a, false, b2,
                                                   (short)0, acc2, false, false);
      acc3 = __builtin_amdgcn_wmma_f32_16x16x4_f32(false, a, false, b3,
                                                   (short)0, acc3, false, false);
    }

    // Own async writes done -> signal; barrier makes them block-visible and
    // confirms everyone finished reading lds[buf] before it is overwritten.
    asm volatile("s_wait_asynccnt 0x0" ::: "memory");
    __syncthreads();
    buf ^= 1;

    if (dch8 == 7) {
      // Fold this 64-center tile into the running per-row argmin.
      // C/D layout: VGPR r, lanes 0-15 -> row r, lanes 16-31 -> row r+8;
      // lane holds column l15 of the 16-wide sub-tile.
      #pragma unroll
      for (int sub = 0; sub < 4; ++sub) {
        const v8f acc = (sub == 0) ? acc0 : (sub == 1) ? acc1
                                          : (sub == 2) ? acc2 : acc3;
        const int col = kt + sub * 16 + l15;
        const float cc = c2[col];
        #pragma unroll
        for (int r = 0; r < 8; ++r) {
          const float dist = fmaf(-2.0f, acc[r], cc);
          if (dist < best[r]) { best[r] = dist; bidx[r] = col; }  // first-min
        }
      }
    }
  }

  // Cross-lane argmin within each 16-lane half (masks <16 stay in-half).
  #pragma unroll
  for (int r = 0; r < 8; ++r) {
    float v = best[r];
    int   i = bidx[r];
    #pragma unroll
    for (int off = 8; off >= 1; off >>= 1) {
      const float ov = __shfl_xor(v, off, 32);
      const int   oi = __shfl_xor(i, off, 32);
      if (ov < v || (ov == v && oi < i)) { v = ov; i = oi; }
    }
    if (l15 == 0) {
      const int row = row0 + r + half * 8;
      out[1 + row] = (float)i;  // indices <= 1023: exact in fp32
    }
  }
}

// ---------------------------------------------------------------------------
// Kernel 3: loss = sum_i (||x_i||^2 + ||c_{y_i}||^2 - 2 x_i . c_{y_i}).
// One wave per row, 4 rows per wave, 32 rows per block, 1 atomic per block.
// ---------------------------------------------------------------------------
__global__ __launch_bounds__(256) void KM_loss(
    const float* __restrict__ x, const int* __restrict__ y,
    const float* __restrict__ centers, float* __restrict__ out) {
  __shared__ float wsum[8];
  const int lane = threadIdx.x & 31;
  const int wave = threadIdx.x >> 5;

  float wacc = 0.0f;
  #pragma unroll
  for (int j = 0; j < 4; ++j) {
    const int i = blockIdx.x * 32 + wave * 4 + j;
    const float* xr = x + (size_t)i * DDIM;
    const float* cr = centers + (size_t)y[i] * DDIM;
    float x2 = 0.0f, cs = 0.0f, dt = 0.0f;
    #pragma unroll 4
    for (int d = lane; d < DDIM; d += 32) {
      const float xv = xr[d];
      const float cv = cr[d];
      x2 = fmaf(xv, xv, x2);
      cs = fmaf(cv, cv, cs);
      dt = fmaf(xv, cv, dt);
    }
    wacc += x2 + cs - 2.0f * dt;
  }
  #pragma unroll
  for (int off = 16; off >= 1; off >>= 1) wacc += __shfl_xor(wacc, off, 32);
  if (lane == 0) wsum[wave] = wacc;
  __syncthreads();
  if (threadIdx.x == 0) {
    float s = 0.0f;
    #pragma unroll
    for (int w = 0; w < 8; ++w) s += wsum[w];
    atomicAdd(out, s);
  }
}

// ---------------------------------------------------------------------------
extern "C" void kernel_launch(void* const* d_in, const int* in_sizes, int n_in,
                              void* d_out, int out_size, void* d_ws, size_t ws_size,
                              hipStream_t stream) {
  const float* x       = (const float*)d_in[0];   // [N, D] fp32
  const int*   y       = (const int*)d_in[1];     // [N]    int32
  const float* centers = (const float*)d_in[2];   // [K, D] fp32
  float* out = (float*)d_out;                     // [1 + N]: loss, ynew
  float* c2  = (float*)d_ws;                      // [K] scratch

  KM_center_norms<<<KCL / 8, 256, 0, stream>>>(centers, c2, out);
  KM_argmin_wmma<<<NPTS / 128, 256, 0, stream>>>(x, centers, c2, out);
  KM_loss<<<NPTS / 32, 256, 0, stream>>>(x, y, centers, out);
}